// GraphElementNetwork_26268019982725
// MI455X (gfx1250) — compile-verified
//
#include <hip/hip_runtime.h>
#include <hip/hip_bf16.h>
#include <math.h>

// ---------------------------------------------------------------------------
// GraphElementNetwork on MI455X (gfx1250, wave32).
//  * Dense MLPs (node encoder 128->64, edge MLP 3(pad4)->64) on the matrix
//    pipe via V_WMMA_F32_16X16X4_F32 (full fp32 precision).
//  * Encoder stages W1 in LDS pair-interleaved (padded stride) so each B
//    fragment is one conflict-free ds_load_b64; grid-stride amortizes staging.
//  * Second layer (64->1) reduced fully in registers with shfl_xor butterflies
//    over the WMMA C-layout (no LDS round trip).
//  * Edge softmax: segment-max via atomicMax on an order-preserving u32 key,
//    then one fused pass accumulating s=Σexp and num=Σexp*h[src]; h = num/s.
//  * Edge-MLP kernel fuses the agg scatter-add and the m2 segment-max.
// Workspace: (10*N + E) floats  (~29.6 MB).
// ---------------------------------------------------------------------------

#define NODE_IN 128
#define HID 64
#define ORD_NEG_INF 0x007FFFFFu   // f2ord(-inf)

typedef __attribute__((ext_vector_type(2))) float v2f;
typedef __attribute__((ext_vector_type(8))) float v8f;

// order-preserving float<->uint map (monotonic: a<b  <=>  f2ord(a)<f2ord(b))
__device__ __forceinline__ unsigned f2ord(float f) {
    unsigned u = __float_as_uint(f);
    return (u & 0x80000000u) ? ~u : (u | 0x80000000u);
}
__device__ __forceinline__ float ord2f(unsigned u) {
    return __uint_as_float((u & 0x80000000u) ? (u & 0x7FFFFFFFu) : ~u);
}

// ---------------- init workspace (ws is poisoned each timing run) ----------
__global__ void k_init(unsigned* m1, unsigned* m2, float* s1, float* num1,
                       float* s2, float* num2, float* agg, int N) {
    int i = blockIdx.x * blockDim.x + threadIdx.x;
    if (i >= N) return;
    m1[i] = ORD_NEG_INF; m2[i] = ORD_NEG_INF;
    s1[i] = 0.f; num1[i] = 0.f; s2[i] = 0.f; num2[i] = 0.f; agg[i] = 0.f;
}

// ---------------- node encoder: relu(relu(X@W1+b1)@w2+b2), WMMA f32 -------
// Block = 128 threads (4 waves). Grid-stride over 16-row tiles.
#define WSTRIDE 80   // float2 stride per k-pair row; +32-dword shift between halves
__global__ __launch_bounds__(128) void k_encoder(
    const float* __restrict__ X, const float* __restrict__ W1,
    const float* __restrict__ b1, const float* __restrict__ w2,
    const float* __restrict__ b2, float* __restrict__ h_enc, int N)
{
    __shared__ v2f lds_w[64 * WSTRIDE];          // 40 KB: pair-interleaved W1
    const int tid = threadIdx.x;
    // stage W1: lds_w[p*WSTRIDE + c] = (W1[2p][c], W1[2p+1][c]); coalesced
    for (int idx = tid; idx < 64 * 64; idx += 128) {
        int p = idx >> 6, c = idx & 63;
        v2f w; w.x = W1[(2 * p) * HID + c]; w.y = W1[(2 * p + 1) * HID + c];
        lds_w[p * WSTRIDE + c] = w;
    }
    __syncthreads();

    const int wid   = tid >> 5;
    const int lane  = tid & 31;
    const int lhalf = lane >> 4;                 // 0: K={0,1}, 1: K={2,3}
    const int l16   = lane & 15;
    const int wtiles = (N + 15) >> 4;

    // per-lane loop-invariant second-layer fragments: col = nt*16 + l16
    float b1v[4], w2v[4];
    #pragma unroll
    for (int nt = 0; nt < 4; ++nt) {
        b1v[nt] = b1[nt * 16 + l16];
        w2v[nt] = w2[nt * 16 + l16];
    }
    const float b2s = b2[0];

    for (int wt = blockIdx.x * 4 + wid; wt < wtiles; wt += gridDim.x * 4) {
        const int row0 = wt * 16;
        int myrow = row0 + l16; if (myrow >= N) myrow = N - 1;

        v8f acc[4] = {};                         // 16x64 hidden accumulator
        const float* arow = X + (size_t)myrow * NODE_IN;
        #pragma unroll 4
        for (int k = 0; k < NODE_IN; k += 4) {
            v2f a = *(const v2f*)(arow + k + 2 * lhalf);      // A[16x4] frag
            const int p = (k >> 1) + lhalf;                   // k-pair index
            #pragma unroll
            for (int nt = 0; nt < 4; ++nt) {
                v2f b = lds_w[p * WSTRIDE + nt * 16 + l16];   // one ds_load_b64
                acc[nt] = __builtin_amdgcn_wmma_f32_16x16x4_f32(
                    false, a, false, b, (short)0, acc[nt], false, false);
            }
        }
        // second layer 64->1: register butterfly over C-layout
        float myval = 0.f;
        #pragma unroll
        for (int v = 0; v < 8; ++v) {
            float partial = 0.f;
            #pragma unroll
            for (int nt = 0; nt < 4; ++nt)
                partial += fmaxf(acc[nt][v] + b1v[nt], 0.f) * w2v[nt];
            partial += __shfl_xor(partial, 1, 32);
            partial += __shfl_xor(partial, 2, 32);
            partial += __shfl_xor(partial, 4, 32);
            partial += __shfl_xor(partial, 8, 32);             // half-wide sum
            float other = __shfl_xor(partial, 16, 32);
            float rl = lhalf ? other : partial;                // row v
            float rh = lhalf ? partial : other;                // row v+8
            if ((l16 & 7) == v) myval = (l16 < 8) ? rl : rh;   // keep own row
        }
        if (lane < 16 && row0 + lane < N)
            h_enc[row0 + lane] = fmaxf(myval + b2s, 0.f);
    }
}

// ---------------- softmax agg #1: segment max of edge_dist -----------------
__global__ void k_max1(const float* __restrict__ dist, const int* __restrict__ dst,
                       unsigned* __restrict__ m1, int E) {
    int e = blockIdx.x * blockDim.x + threadIdx.x;
    if (e >= E) return;
    atomicMax(&m1[dst[e]], f2ord(dist[e]));
}

// fused: s += exp(l-m),  num += exp(l-m)*h[src]
__global__ void k_acc1(const float* __restrict__ dist, const int* __restrict__ src,
                       const int* __restrict__ dst, const unsigned* __restrict__ m1,
                       const float* __restrict__ h, float* __restrict__ s1,
                       float* __restrict__ num1, int E) {
    int e = blockIdx.x * blockDim.x + threadIdx.x;
    if (e >= E) return;
    int d = dst[e];
    float ex = expf(dist[e] - ord2f(m1[d]));
    atomicAdd(&s1[d], ex);
    atomicAdd(&num1[d], ex * h[src[e]]);
}

__global__ void k_div(const float* __restrict__ s, const float* __restrict__ num,
                      float* __restrict__ h, int N) {
    int i = blockIdx.x * blockDim.x + threadIdx.x;
    if (i >= N) return;
    float sv = s[i];
    h[i] = (sv > 0.f) ? num[i] / sv : 0.f;       // empty segment -> 0 (matches ref)
}

// ---------------- edge MLP (WMMA, K=3 padded to 4) + fused agg/max2 --------
// Block = 256 threads (8 waves); grid-stride, 16 edges per wave-tile.
__global__ __launch_bounds__(256) void k_edge_mlp(
    const float* __restrict__ ef, const int* __restrict__ src,
    const int* __restrict__ dst, const float* __restrict__ h1,
    const float* __restrict__ W1, const float* __restrict__ b1,
    const float* __restrict__ w2, const float* __restrict__ b2,
    float* __restrict__ eh, float* __restrict__ agg,
    unsigned* __restrict__ m2, int E)
{
    const int wid   = threadIdx.x >> 5;
    const int lane  = threadIdx.x & 31;
    const int lhalf = lane >> 4;
    const int l16   = lane & 15;
    const int tiles = (E + 15) >> 4;

    // B frag: rows = [w_ef; w_hs; w_hd; 0]  (loop invariant)
    v2f bfr[4];
    #pragma unroll
    for (int nt = 0; nt < 4; ++nt) {
        if (lhalf == 0) { bfr[nt].x = W1[      nt * 16 + l16];
                          bfr[nt].y = W1[ 64 + nt * 16 + l16]; }
        else            { bfr[nt].x = W1[128 + nt * 16 + l16];
                          bfr[nt].y = 0.f; }
    }
    float b1v[4], w2v[4];
    #pragma unroll
    for (int nt = 0; nt < 4; ++nt) {
        b1v[nt] = b1[nt * 16 + l16];
        w2v[nt] = w2[nt * 16 + l16];
    }
    const float b2s = b2[0];

    for (int tile = blockIdx.x * 8 + wid; tile < tiles; tile += gridDim.x * 8) {
        const int base = tile * 16;
        int eid = base + l16; if (eid >= E) eid = E - 1;

        // A frag: lanes 0-15 -> (ef, h[src]); lanes 16-31 -> (h[dst], 0)
        v2f a; int gidx;
        if (lhalf == 0) { gidx = src[eid]; a.x = ef[eid]; a.y = h1[gidx]; }
        else            { gidx = dst[eid]; a.x = h1[gidx]; a.y = 0.f; }

        v8f acc[4] = {};
        #pragma unroll
        for (int nt = 0; nt < 4; ++nt)
            acc[nt] = __builtin_amdgcn_wmma_f32_16x16x4_f32(
                false, a, false, bfr[nt], (short)0, acc[nt], false, false);

        float myval = 0.f;
        #pragma unroll
        for (int v = 0; v < 8; ++v) {
            float partial = 0.f;
            #pragma unroll
            for (int nt = 0; nt < 4; ++nt)
                partial += fmaxf(acc[nt][v] + b1v[nt], 0.f) * w2v[nt];
            partial += __shfl_xor(partial, 1, 32);
            partial += __shfl_xor(partial, 2, 32);
            partial += __shfl_xor(partial, 4, 32);
            partial += __shfl_xor(partial, 8, 32);
            float other = __shfl_xor(partial, 16, 32);
            float rl = lhalf ? other : partial;
            float rh = lhalf ? partial : other;
            if ((l16 & 7) == v) myval = (l16 < 8) ? rl : rh;
        }
        int didx = __shfl_xor(gidx, 16, 32);   // lane L<16 gets dst[base+L]
        if (lane < 16 && base + lane < E) {
            float ev = fmaxf(myval + b2s, 0.f);
            eh[base + lane] = ev;
            atomicAdd(&agg[didx], ev);          // pull: segment_sum(eh, dst)
            atomicMax(&m2[didx], f2ord(ev));    // segment_max for softmax #2
        }
    }
}

// ---------------- node update MLP: [agg, h1] -> 64 -> 1 --------------------
__global__ void k_nodeupd(const float* __restrict__ agg, const float* __restrict__ h1,
                          const float* __restrict__ W1, const float* __restrict__ b1,
                          const float* __restrict__ w2, const float* __restrict__ b2,
                          float* __restrict__ h2, int N) {
    int i = blockIdx.x * blockDim.x + threadIdx.x;
    if (i >= N) return;
    float x0 = agg[i], x1 = h1[i];
    float o = b2[0];
    #pragma unroll 8
    for (int j = 0; j < HID; ++j)
        o += fmaxf(W1[j] * x0 + W1[HID + j] * x1 + b1[j], 0.f) * w2[j];
    h2[i] = fmaxf(o, 0.f);
}

// ---------------- softmax agg #2 accumulate --------------------------------
__global__ void k_acc2(const float* __restrict__ eh, const int* __restrict__ src,
                       const int* __restrict__ dst, const unsigned* __restrict__ m2,
                       const float* __restrict__ h2, float* __restrict__ s2,
                       float* __restrict__ num2, int E) {
    int e = blockIdx.x * blockDim.x + threadIdx.x;
    if (e >= E) return;
    int d = dst[e];
    float ex = expf(eh[e] - ord2f(m2[d]));
    atomicAdd(&s2[d], ex);
    atomicAdd(&num2[d], ex * h2[src[e]]);
}

// ---------------- finalize: softmax divide + decoder MLP 1->64->1 ----------
__global__ void k_final(const float* __restrict__ s2, const float* __restrict__ num2,
                        const float* __restrict__ W1, const float* __restrict__ b1,
                        const float* __restrict__ w2, const float* __restrict__ b2,
                        float* __restrict__ out, int N) {
    int i = blockIdx.x * blockDim.x + threadIdx.x;
    if (i >= N) return;
    float sv = s2[i];
    float hv = (sv > 0.f) ? num2[i] / sv : 0.f;
    float o = b2[0];
    #pragma unroll 8
    for (int j = 0; j < HID; ++j)
        o += fmaxf(W1[j] * hv + b1[j], 0.f) * w2[j];
    out[i] = fmaxf(o, 0.f);
}

// ---------------------------------------------------------------------------
extern "C" void kernel_launch(void* const* d_in, const int* in_sizes, int n_in,
                              void* d_out, int out_size, void* d_ws, size_t ws_size,
                              hipStream_t stream) {
    const float* node_feat = (const float*)d_in[0];
    const float* edge_feat = (const float*)d_in[1];
    const float* edge_dist = (const float*)d_in[2];
    const int*   src       = (const int*)  d_in[3];
    const int*   dst       = (const int*)  d_in[4];
    const float* enc_w1 = (const float*)d_in[5],  *enc_b1 = (const float*)d_in[6];
    const float* enc_w2 = (const float*)d_in[7],  *enc_b2 = (const float*)d_in[8];
    const float* nu_w1  = (const float*)d_in[9],  *nu_b1  = (const float*)d_in[10];
    const float* nu_w2  = (const float*)d_in[11], *nu_b2  = (const float*)d_in[12];
    const float* eu_w1  = (const float*)d_in[13], *eu_b1  = (const float*)d_in[14];
    const float* eu_w2  = (const float*)d_in[15], *eu_b2  = (const float*)d_in[16];
    const float* dec_w1 = (const float*)d_in[17], *dec_b1 = (const float*)d_in[18];
    const float* dec_w2 = (const float*)d_in[19], *dec_b2 = (const float*)d_in[20];

    const int N = in_sizes[0] / NODE_IN;   // 100000
    const int E = in_sizes[3];             // 6400000

    // workspace layout: (10*N + E) floats
    float*    h_enc = (float*)d_ws;
    unsigned* m1    = (unsigned*)(h_enc + N);
    float*    s1    = (float*)(m1 + N);
    float*    num1  = s1 + N;
    float*    h1    = num1 + N;
    float*    agg   = h1 + N;
    unsigned* m2    = (unsigned*)(agg + N);
    float*    s2    = (float*)(m2 + N);
    float*    num2  = s2 + N;
    float*    h2    = num2 + N;
    float*    eh    = h2 + N;              // E floats

    const int nbN = (N + 255) / 256;
    const int nbE = (E + 255) / 256;

    k_init<<<nbN, 256, 0, stream>>>(m1, m2, s1, num1, s2, num2, agg, N);

    // encoder: 512 blocks x 4 waves, grid-stride over 16-row tiles
    const int wtiles = (N + 15) / 16;
    int enc_blocks = (wtiles + 3) / 4; if (enc_blocks > 512) enc_blocks = 512;
    k_encoder<<<enc_blocks, 128, 0, stream>>>(
        node_feat, enc_w1, enc_b1, enc_w2, enc_b2, h_enc, N);

    k_max1<<<nbE, 256, 0, stream>>>(edge_dist, dst, m1, E);
    k_acc1<<<nbE, 256, 0, stream>>>(edge_dist, src, dst, m1, h_enc, s1, num1, E);
    k_div <<<nbN, 256, 0, stream>>>(s1, num1, h1, N);

    // edge MLP: grid-stride, ~4 tiles per wave
    const int etiles = (E + 15) / 16;
    int edge_blocks = (etiles + 8 * 4 - 1) / (8 * 4);
    k_edge_mlp<<<edge_blocks, 256, 0, stream>>>(
        edge_feat, src, dst, h1, eu_w1, eu_b1, eu_w2, eu_b2, eh, agg, m2, E);

    k_nodeupd<<<nbN, 256, 0, stream>>>(agg, h1, nu_w1, nu_b1, nu_w2, nu_b2, h2, N);
    k_acc2   <<<nbE, 256, 0, stream>>>(eh, src, dst, m2, h2, s2, num2, E);
    k_final  <<<nbN, 256, 0, stream>>>(s2, num2, dec_w1, dec_b1, dec_w2, dec_b2,
                                       (float*)d_out, N);
}